// Despawn2D_41901700940346
// MI455X (gfx1250) — compile-verified
//
#include <hip/hip_runtime.h>

// CDNA5 (gfx1250) wave32 WMMA implementation of Despawn2D:
// 6-level separable circular db4 DWT (fp32), sigmoid thresholding, inverse DWT.
// Convolutions are banded 16x16 tile matmuls on V_WMMA_F32_16X16X4_F32.
// - Banded-filter fragments precomputed once into a small table.
// - Hot loops use uniform-SADDR + single per-lane voffset addressing.
// - Circular wrap handled in a scalar-branch slow path taken only by the
//   first/last tile blocks (branch on blockIdx -> s_cbranch, EXEC stays ~0xFFFFFFFF).

typedef __attribute__((ext_vector_type(2))) float v2f;
typedef __attribute__((ext_vector_type(8))) float v8f;

__device__ __forceinline__ v8f wmma_f32(v2f a, v2f b, v8f c) {
    // 8 args: (neg_a, A, neg_b, B, c_mod, C, reuse_a, reuse_b)
    return __builtin_amdgcn_wmma_f32_16x16x4_f32(false, a, false, b, (short)0, c,
                                                 false, false);
}

__device__ __forceinline__ float ftap(const float f[8], int j) {
    float r = 0.0f;
#pragma unroll
    for (int t = 0; t < 8; ++t) r = (j == t) ? f[t] : r;
    return r;
}

__device__ __forceinline__ void load_filters(const float* __restrict__ hrow,
                                             float h[8], float w[8]) {
#pragma unroll
    for (int j = 0; j < 8; ++j) h[j] = hrow[j];
    // w[j] = (-1)^j * h[7-j]   (flip + negate odd indices)
#pragma unroll
    for (int j = 0; j < 8; ++j) w[j] = (j & 1) ? -h[7 - j] : h[7 - j];
}

__device__ __forceinline__ float thrf(float t, float alpha, float bp, float bm) {
    float s1 = 1.0f / (1.0f + __expf(-alpha * (t - bp)));
    float s2 = 1.0f / (1.0f + __expf(alpha * (t + bm)));
    return t * (s1 + s2);
}

// Table layout (v2f units):
//   FWDW[7][10][32], FWDH[7][10][32]   fwd band: f[2*lid + 7 - k]
//   BWDW[7][3][32],  BWDH[7][3][32]    bwd band: f[2*k - lid]
#define FWD_STRIDE (10 * 32)
#define BWD_STRIDE (3 * 32)

__global__ void __launch_bounds__(256)
build_frag_tables(const float* __restrict__ scaling, float* __restrict__ tabf) {
    int idx = blockIdx.x * blockDim.x + threadIdx.x;
    if (idx >= 7 * 10 * 32) return;
    int lane = idx & 31;
    int st = (idx >> 5) % 10;
    int r = idx / 320;
    float h[8], w[8];
    load_filters(scaling + 8 * r, h, w);
    int half = lane >> 4, lid = lane & 15;
    int k0 = 4 * st + 2 * half;
    v2f* FWDW = (v2f*)tabf;
    v2f* FWDH = FWDW + 7 * FWD_STRIDE;
    v2f* BWDW = FWDH + 7 * FWD_STRIDE;
    v2f* BWDH = BWDW + 7 * BWD_STRIDE;
    int j0 = 2 * lid + 7 - k0;
    v2f e;
    e.x = ftap(w, j0);  e.y = ftap(w, j0 - 1);
    FWDW[r * FWD_STRIDE + st * 32 + lane] = e;
    e.x = ftap(h, j0);  e.y = ftap(h, j0 - 1);
    FWDH[r * FWD_STRIDE + st * 32 + lane] = e;
    if (st < 3) {
        e.x = ftap(w, 2 * k0 - lid);  e.y = ftap(w, 2 * k0 + 2 - lid);
        BWDW[r * BWD_STRIDE + st * 32 + lane] = e;
        e.x = ftap(h, 2 * k0 - lid);  e.y = ftap(h, 2 * k0 + 2 - lid);
        BWDH[r * BWD_STRIDE + st * 32 + lane] = e;
    }
}

// -------------------- Forward row pass --------------------
// d[r][i] = sum_j w[j] x[r][(2i-j) mod S]; D(16x16) = X(16x40)*W(40x16 banded).
__global__ void __launch_bounds__(128)
fwd_row_wmma(const float* __restrict__ Ain, float* __restrict__ Td,
             float* __restrict__ Ta, const v2f* __restrict__ tabW,
             const v2f* __restrict__ tabH, int S) {
    const int lane = threadIdx.x & 31;
    const int tc = blockIdx.x * 4 + (threadIdx.x >> 5);
    const int tr = blockIdx.y;
    const int outc = S >> 1;
    const int r0 = tr << 4, n0 = tc << 4;
    const int mask = S - 1;
    const int half = lane >> 4, lid = lane & 15;

    v8f Cd = {}, Ca = {};
    const int base = ((n0 << 1) - 7 + S) & mask;   // 2*n0 - 7 mod S
    // scalar branch: only first/last tile-column blocks can wrap
    if (blockIdx.x != 0 && blockIdx.x != gridDim.x - 1) {
        const int rowoff = (r0 + lid) * S + base + 2 * half;  // loop-invariant voffset
        __builtin_prefetch(Ain + rowoff, 0, 0);               // global_prefetch_b8
#pragma unroll
        for (int st = 0; st < 10; ++st) {
            v2f Bw = tabW[st * 32 + lane];
            v2f Bh = tabH[st * 32 + lane];
            v2f A;
            A.x = Ain[rowoff + 4 * st];       // imm-offset folded
            A.y = Ain[rowoff + 4 * st + 1];
            Cd = wmma_f32(A, Bw, Cd);
            Ca = wmma_f32(A, Bh, Ca);
        }
    } else {
        const float* rowp = Ain + (r0 + lid) * S;
#pragma unroll
        for (int st = 0; st < 10; ++st) {
            v2f Bw = tabW[st * 32 + lane];
            v2f Bh = tabH[st * 32 + lane];
            int k0 = 4 * st + 2 * half;
            v2f A;
            A.x = rowp[(base + k0) & mask];
            A.y = rowp[(base + k0 + 1) & mask];
            Cd = wmma_f32(A, Bw, Cd);
            Ca = wmma_f32(A, Bh, Ca);
        }
    }
    const int soff = 8 * half * outc + n0 + lid;  // loop-invariant voffset
#pragma unroll
    for (int v = 0; v < 8; ++v) {
        float* td = Td + (r0 + v) * outc;         // uniform SALU pointer
        float* ta = Ta + (r0 + v) * outc;
        td[soff] = Cd[v];
        ta[soff] = Ca[v];
    }
}

// -------------------- Forward column pass + threshold --------------------
__global__ void __launch_bounds__(128)
fwd_col_thr_wmma(const float* __restrict__ Td, const float* __restrict__ Ta,
                 int S, float* __restrict__ flat, float* __restrict__ Anext,
                 const v2f* __restrict__ tabW, const v2f* __restrict__ tabH,
                 const float* __restrict__ alphaP, const float* __restrict__ bpP,
                 const float* __restrict__ bmP) {
    const int lane = threadIdx.x & 31;
    const int tc = blockIdx.x * 4 + (threadIdx.x >> 5);
    const int ti = blockIdx.y;
    const int s = S >> 1;
    const int i0 = ti << 4, c0 = tc << 4;
    const int half = lane >> 4, lid = lane & 15;

    const float alpha = alphaP[0], bp = bpP[0], bm = bmP[0];

    v8f Chh = {}, Chl = {}, Clh = {}, Cll = {};
    const int rbase = ((i0 << 1) - 7 + S) & (S - 1);
    const int lidoff = 2 * half * s + c0 + lid;   // loop-invariant voffset
    const int lidoff2 = lidoff + s;
    // scalar branch: wrap only in first/last tile-row blocks (blockIdx.y)
    if (blockIdx.y != 0 && blockIdx.y != gridDim.y - 1) {
#pragma unroll
        for (int st = 0; st < 10; ++st) {
            v2f Aw = tabW[st * 32 + lane];
            v2f Ah = tabH[st * 32 + lane];
            const int u = (rbase + 4 * st) * s;   // uniform SALU
            const float* pd = Td + u;
            const float* pa = Ta + u;
            v2f Bd, Ba;
            Bd.x = pd[lidoff];   Bd.y = pd[lidoff2];
            Ba.x = pa[lidoff];   Ba.y = pa[lidoff2];
            Chh = wmma_f32(Aw, Bd, Chh);
            Chl = wmma_f32(Ah, Bd, Chl);
            Clh = wmma_f32(Aw, Ba, Clh);
            Cll = wmma_f32(Ah, Ba, Cll);
        }
    } else {
#pragma unroll
        for (int st = 0; st < 10; ++st) {
            v2f Aw = tabW[st * 32 + lane];
            v2f Ah = tabH[st * 32 + lane];
            int k0 = 4 * st + 2 * half;
            int ra = (rbase + k0) & (S - 1);
            int rb = (rbase + k0 + 1) & (S - 1);
            v2f Bd, Ba;
            Bd.x = Td[ra * s + c0 + lid];
            Bd.y = Td[rb * s + c0 + lid];
            Ba.x = Ta[ra * s + c0 + lid];
            Ba.y = Ta[rb * s + c0 + lid];
            Chh = wmma_f32(Aw, Bd, Chh);
            Chl = wmma_f32(Ah, Bd, Chl);
            Clh = wmma_f32(Aw, Ba, Clh);
            Cll = wmma_f32(Ah, Ba, Cll);
        }
    }
    const int foff = 8 * half * (3 * s) + c0 + lid;
    const int aoff = 8 * half * s + c0 + lid;
#pragma unroll
    for (int v = 0; v < 8; ++v) {
        float* fr = flat + (i0 + v) * (3 * s);    // uniform SALU pointer
        float* an = Anext + (i0 + v) * s;
        fr[foff]         = thrf(Chh[v], alpha, bp, bm);
        fr[foff + s]     = thrf(Chl[v], alpha, bp, bm);
        fr[foff + 2 * s] = thrf(Clh[v], alpha, bp, bm);
        an[aoff]         = Cll[v];
    }
}

// -------------------- Threshold final approx (64x64) --------------------
__global__ void __launch_bounds__(256)
thr_approx(float* __restrict__ A, float* __restrict__ flat,
           const float* __restrict__ alphaP, const float* __restrict__ bpP,
           const float* __restrict__ bmP) {
    int i = blockIdx.x * blockDim.x + threadIdx.x;
    if (i >= 64 * 64) return;
    float v = thrf(A[i], alphaP[0], bpP[0], bmP[0]);
    A[i] = v;        // backward pass reads this
    flat[i] = v;     // flat output tail
}

// -------------------- Backward column pass --------------------
// h[n][c] = sum_j w[j]*up(hh)[(n+j)%S][c] + h[j]*up(hl)[...]; band K=12.
__global__ void __launch_bounds__(128)
bwd_col_wmma(const float* __restrict__ flat, const float* __restrict__ Aap,
             float* __restrict__ Th, float* __restrict__ Tl, int S,
             const v2f* __restrict__ tabW, const v2f* __restrict__ tabH) {
    const int lane = threadIdx.x & 31;
    const int tc = blockIdx.x * 4 + (threadIdx.x >> 5);
    const int tn = blockIdx.y;
    const int s = S >> 1;
    const int n0 = tn << 4, c0 = tc << 4;
    const int half = lane >> 4, lid = lane & 15;
    const int smask = s - 1;

    v8f Ch = {}, Cl = {};
    const int tbase = n0 >> 1;
    const int lidf = 2 * half * (3 * s) + c0 + lid;   // flat voffset
    const int lidf2 = lidf + 3 * s;
    const int lida = 2 * half * s + c0 + lid;         // approx voffset
    const int lida2 = lida + s;
    // scalar branch: wrap only in last tile-row block (tbase never negative)
    if (blockIdx.y != gridDim.y - 1) {
#pragma unroll
        for (int st = 0; st < 3; ++st) {
            v2f Aw = tabW[st * 32 + lane];
            v2f Ah = tabH[st * 32 + lane];
            const int uf = (tbase + 4 * st) * (3 * s);  // uniform SALU
            const int ua = (tbase + 4 * st) * s;
            const float* pf = flat + uf;
            const float* pa = Aap + ua;
            v2f Bhh, Bhl, Blh, Ba;
            Bhh.x = pf[lidf];          Bhh.y = pf[lidf2];
            Bhl.x = pf[lidf + s];      Bhl.y = pf[lidf2 + s];
            Blh.x = pf[lidf + 2 * s];  Blh.y = pf[lidf2 + 2 * s];
            Ba.x  = pa[lida];          Ba.y  = pa[lida2];
            Ch = wmma_f32(Aw, Bhh, Ch);
            Ch = wmma_f32(Ah, Bhl, Ch);
            Cl = wmma_f32(Aw, Blh, Cl);
            Cl = wmma_f32(Ah, Ba,  Cl);
        }
    } else {
#pragma unroll
        for (int st = 0; st < 3; ++st) {
            v2f Aw = tabW[st * 32 + lane];
            v2f Ah = tabH[st * 32 + lane];
            int k0 = 4 * st + 2 * half;
            int ra = (tbase + k0) & smask;
            int rb = (tbase + k0 + 1) & smask;
            int fa = ra * (3 * s) + c0 + lid;
            int fb = rb * (3 * s) + c0 + lid;
            v2f Bhh, Bhl, Blh, Ba;
            Bhh.x = flat[fa];          Bhh.y = flat[fb];
            Bhl.x = flat[fa + s];      Bhl.y = flat[fb + s];
            Blh.x = flat[fa + 2 * s];  Blh.y = flat[fb + 2 * s];
            Ba.x  = Aap[ra * s + c0 + lid];
            Ba.y  = Aap[rb * s + c0 + lid];
            Ch = wmma_f32(Aw, Bhh, Ch);
            Ch = wmma_f32(Ah, Bhl, Ch);
            Cl = wmma_f32(Aw, Blh, Cl);
            Cl = wmma_f32(Ah, Ba,  Cl);
        }
    }
    const int soff = 8 * half * s + c0 + lid;
#pragma unroll
    for (int v = 0; v < 8; ++v) {
        float* th = Th + (n0 + v) * s;            // uniform SALU pointer
        float* tl = Tl + (n0 + v) * s;
        th[soff] = Ch[v];
        tl[soff] = Cl[v];
    }
}

// -------------------- Backward row pass --------------------
__global__ void __launch_bounds__(128)
bwd_row_wmma(const float* __restrict__ Th, const float* __restrict__ Tl,
             float* __restrict__ Out, int S,
             const v2f* __restrict__ tabW, const v2f* __restrict__ tabH) {
    const int lane = threadIdx.x & 31;
    const int tx = blockIdx.x * 4 + (threadIdx.x >> 5);
    const int tr = blockIdx.y;
    const int r0 = tr << 4, x0 = tx << 4;
    const int s = S >> 1, smask = s - 1;
    const int half = lane >> 4, lid = lane & 15;

    v8f C = {};
    const int tbase = x0 >> 1;
    // scalar branch: wrap only in last tile-column block
    if (blockIdx.x != gridDim.x - 1) {
        const int rowoff = (r0 + lid) * s + tbase + 2 * half;  // voffset
#pragma unroll
        for (int st = 0; st < 3; ++st) {
            v2f Bw = tabW[st * 32 + lane];
            v2f Bh = tabH[st * 32 + lane];
            v2f A1, A2;
            A1.x = Th[rowoff + 4 * st];   A1.y = Th[rowoff + 4 * st + 1];
            A2.x = Tl[rowoff + 4 * st];   A2.y = Tl[rowoff + 4 * st + 1];
            C = wmma_f32(A1, Bw, C);
            C = wmma_f32(A2, Bh, C);
        }
    } else {
        const float* rh = Th + (r0 + lid) * s;
        const float* rl = Tl + (r0 + lid) * s;
#pragma unroll
        for (int st = 0; st < 3; ++st) {
            v2f Bw = tabW[st * 32 + lane];
            v2f Bh = tabH[st * 32 + lane];
            int k0 = 4 * st + 2 * half;
            int ca = (tbase + k0) & smask;
            int cb = (tbase + k0 + 1) & smask;
            v2f A1, A2;
            A1.x = rh[ca];  A1.y = rh[cb];
            A2.x = rl[ca];  A2.y = rl[cb];
            C = wmma_f32(A1, Bw, C);
            C = wmma_f32(A2, Bh, C);
        }
    }
    const int soff = 8 * half * S + x0 + lid;
#pragma unroll
    for (int v = 0; v < 8; ++v) {
        float* o = Out + (r0 + v) * S;            // uniform SALU pointer
        o[soff] = C[v];
    }
}

// -------------------- Host orchestration --------------------
extern "C" void kernel_launch(void* const* d_in, const int* in_sizes, int n_in,
                              void* d_out, int out_size, void* d_ws, size_t ws_size,
                              hipStream_t stream) {
    const float* x       = (const float*)d_in[0];
    const float* scaling = (const float*)d_in[1];   // (12, 8)
    const float* alpha   = (const float*)d_in[2];
    const float* bp      = (const float*)d_in[3];
    const float* bm      = (const float*)d_in[4];
    float* out = (float*)d_out;

    const int N = 4096, LEVELS = 6;
    float* T  = (float*)d_ws;                              // N*N floats (temps)
    float* P0 = T + (size_t)N * N;                         // (N/2)^2 ping
    float* P1 = P0 + (size_t)(N / 2) * (N / 2);            // (N/2)^2 pong
    float* TABF = P1 + (size_t)(N / 2) * (N / 2);          // fragment tables
    v2f* FWDW = (v2f*)TABF;
    v2f* FWDH = FWDW + 7 * FWD_STRIDE;
    v2f* BWDW = FWDH + 7 * FWD_STRIDE;
    v2f* BWDH = BWDW + 7 * BWD_STRIDE;

    build_frag_tables<<<dim3(9), dim3(256), 0, stream>>>(scaling, TABF);

    size_t flatOff[LEVELS + 1];
    flatOff[0] = (size_t)N * N;
    for (int l = 0; l < LEVELS; ++l) {
        int s = N >> (l + 1);
        flatOff[l + 1] = flatOff[l] + (size_t)3 * s * s;
    }

    float* P[2] = {P0, P1};
    const float* cur = x;
    for (int lev = 0; lev < LEVELS; ++lev) {
        int S = N >> lev, s = S >> 1;
        float* Td = T;
        float* Ta = T + (size_t)S * s;
        fwd_row_wmma<<<dim3((s / 16) / 4, S / 16), dim3(128), 0, stream>>>(
            cur, Td, Ta, FWDW + lev * FWD_STRIDE, FWDH + lev * FWD_STRIDE, S);
        float* Anext = P[lev & 1];
        fwd_col_thr_wmma<<<dim3((s / 16) / 4, s / 16), dim3(128), 0, stream>>>(
            Td, Ta, S, out + flatOff[lev], Anext,
            FWDW + (lev + 1) * FWD_STRIDE, FWDH + (lev + 1) * FWD_STRIDE,
            alpha, bp, bm);
        cur = Anext;
    }

    // cur = P1 holds 64x64 final approx
    thr_approx<<<dim3(16), dim3(256), 0, stream>>>(
        (float*)cur, out + flatOff[LEVELS], alpha, bp, bm);

    float* a = (float*)cur;
    for (int lev = LEVELS - 1; lev >= 0; --lev) {
        int s = N >> (lev + 1), S = 2 * s;
        float* Th = T;
        float* Tl = T + (size_t)S * s;
        bwd_col_wmma<<<dim3((s / 16) / 4, S / 16), dim3(128), 0, stream>>>(
            out + flatOff[lev], a, Th, Tl, S,
            BWDW + (lev + 1) * BWD_STRIDE, BWDH + (lev + 1) * BWD_STRIDE);
        float* dst = (lev == 0) ? out : ((a == P0) ? P1 : P0);
        bwd_row_wmma<<<dim3((S / 16) / 4, S / 16), dim3(128), 0, stream>>>(
            Th, Tl, dst, S, BWDW + lev * BWD_STRIDE, BWDH + lev * BWD_STRIDE);
        a = dst;
    }
}